// WeightedChildSumTreeLSTMEndtoEnd_77352361001540
// MI455X (gfx1250) — compile-verified
//
#include <hip/hip_runtime.h>
#include <math.h>

// ---------------------------------------------------------------------------
// Weighted child-sum Tree-LSTM, end to end, fp32 on CDNA5 (gfx1250).
// All heavy math mapped to V_WMMA_F32_16X16X4_F32 (full-precision f32 WMMA).
// ---------------------------------------------------------------------------

typedef __attribute__((ext_vector_type(2))) float v2f;
typedef __attribute__((ext_vector_type(8))) float v8f;

#define N_NODES 585
#define BRANCH  8
#define BATCH   256
#define MEM     256
#define INDIM   512
#define BM      (BATCH * MEM)

__device__ __forceinline__ float sigmoid_f(float x) {
    return 1.0f / (1.0f + __expf(-x));
}

#define WMMA_F32(A, B, C) \
    __builtin_amdgcn_wmma_f32_16x16x4_f32(false, (A), false, (B), (short)0, (C), false, false)

// ---------------------------------------------------------------------------
// hsum[nl][b][m] = sum_k prob[node, child_k] * h_all[child_k][b][m]
// ---------------------------------------------------------------------------
__global__ void hsum_kernel(const float* __restrict__ h_all,
                            const float* __restrict__ prob,
                            float* __restrict__ hsum,
                            int s, int L)
{
    int idx = blockIdx.x * blockDim.x + threadIdx.x;
    int total = L * BM;
    if (idx >= total) return;
    int nl = idx / BM;
    int bm = idx - nl * BM;
    int node = s + nl;
    int child0 = node * BRANCH + 1;
    float acc = 0.0f;
#pragma unroll
    for (int k = 0; k < BRANCH; ++k) {
        int ch = child0 + k;
        float w = prob[node * N_NODES + ch];
        acc += w * h_all[(size_t)ch * BM + bm];
    }
    hsum[(size_t)nl * BM + bm] = acc;
}

// ---------------------------------------------------------------------------
// One level of the tree. Grid: (L, BATCH/64, MEM/64), 512 threads = 16 waves.
// Each wave owns one 16x16 (rows-of-batch x cols-of-MEM) output tile and
// carries 12 f32 WMMA accumulators: i, o, u, fx, fh[0..7].
// The per-child probability weight is applied to the fh accumulator in the
// epilogue (GEMM is linear in A), keeping the hot K-loop pure load->wmma.
// ---------------------------------------------------------------------------
__global__ __launch_bounds__(512)
void tree_lstm_level_kernel(const float* __restrict__ inputs,
                            const float* __restrict__ prob,
                            const float* __restrict__ Wioux,
                            const float* __restrict__ bioux,
                            const float* __restrict__ Wiouh,
                            const float* __restrict__ biouh,
                            const float* __restrict__ Wfx,
                            const float* __restrict__ bfx,
                            const float* __restrict__ Wfh,
                            const float* __restrict__ bfh,
                            const float* __restrict__ hsum,
                            float* __restrict__ c_all,
                            float* __restrict__ h_all,
                            int s, int has_children)
{
    const int nl   = blockIdx.x;
    const int node = s + nl;
    const int tid  = threadIdx.x;
    const int wave = tid >> 5;
    const int lane = tid & 31;
    const int l16  = lane & 15;        // N column (B/C/D) or M row (A) within tile
    const int lhi  = lane >> 4;        // selects K pair {0,1} vs {2,3}
    const int wr   = wave >> 2;        // 0..3 row sub-tile
    const int wc   = wave & 3;         // 0..3 col sub-tile
    const int rowbase = blockIdx.y * 64 + wr * 16;   // batch rows
    const int colbase = blockIdx.z * 64 + wc * 16;   // MEM cols
    const int koff = 2 * lhi;

    v8f acc_i = {};
    v8f acc_o = {};
    v8f acc_u = {};
    v8f acc_fx = {};
    v8f acc_fh[BRANCH] = {};

    // ---------------- x-driven GEMMs (K over INDIM=512) -------------------
    const float* Ax  = inputs + (size_t)node * BATCH * INDIM
                              + (size_t)(rowbase + l16) * INDIM;
    const float* Bxi = Wioux + (size_t)(colbase + l16) * INDIM;
    const float* Bxo = Wioux + (size_t)(colbase + 256 + l16) * INDIM;
    const float* Bxu = Wioux + (size_t)(colbase + 512 + l16) * INDIM;
    const float* Bxf = Wfx   + (size_t)(colbase + l16) * INDIM;

    if (has_children) {
#pragma unroll 2
        for (int k0 = 0; k0 < INDIM; k0 += 4) {
            v2f a  = *(const v2f*)(Ax  + k0 + koff);
            v2f bi = *(const v2f*)(Bxi + k0 + koff);
            v2f bo = *(const v2f*)(Bxo + k0 + koff);
            v2f bu = *(const v2f*)(Bxu + k0 + koff);
            v2f bf = *(const v2f*)(Bxf + k0 + koff);
            acc_i  = WMMA_F32(a, bi, acc_i);
            acc_o  = WMMA_F32(a, bo, acc_o);
            acc_u  = WMMA_F32(a, bu, acc_u);
            acc_fx = WMMA_F32(a, bf, acc_fx);
        }
    } else {
        // Leaves: f-gates multiply zero child_c, so fx/fh are dead. Skip them.
#pragma unroll 2
        for (int k0 = 0; k0 < INDIM; k0 += 4) {
            v2f a  = *(const v2f*)(Ax  + k0 + koff);
            v2f bi = *(const v2f*)(Bxi + k0 + koff);
            v2f bo = *(const v2f*)(Bxo + k0 + koff);
            v2f bu = *(const v2f*)(Bxu + k0 + koff);
            acc_i = WMMA_F32(a, bi, acc_i);
            acc_o = WMMA_F32(a, bo, acc_o);
            acc_u = WMMA_F32(a, bu, acc_u);
        }
    }

    // ---------------- h-driven GEMMs (K over MEM=256) ---------------------
    const int child0 = node * BRANCH + 1;
    float wgt[BRANCH];
#pragma unroll
    for (int k = 0; k < BRANCH; ++k) wgt[k] = 0.0f;

    if (has_children) {
#pragma unroll
        for (int k = 0; k < BRANCH; ++k)
            wgt[k] = prob[node * N_NODES + child0 + k];

        const float* Ah = hsum + (size_t)nl * BM + (size_t)(rowbase + l16) * MEM;
        const float* Hk[BRANCH];
#pragma unroll
        for (int k = 0; k < BRANCH; ++k)
            Hk[k] = h_all + (size_t)(child0 + k) * BM + (size_t)(rowbase + l16) * MEM;

        const float* Bhi = Wiouh + (size_t)(colbase + l16) * MEM;
        const float* Bho = Wiouh + (size_t)(colbase + 256 + l16) * MEM;
        const float* Bhu = Wiouh + (size_t)(colbase + 512 + l16) * MEM;
        const float* Bfh = Wfh   + (size_t)(colbase + l16) * MEM;

#pragma unroll 2
        for (int k0 = 0; k0 < MEM; k0 += 4) {
            v2f a  = *(const v2f*)(Ah  + k0 + koff);
            v2f bi = *(const v2f*)(Bhi + k0 + koff);
            v2f bo = *(const v2f*)(Bho + k0 + koff);
            v2f bu = *(const v2f*)(Bhu + k0 + koff);
            v2f bf = *(const v2f*)(Bfh + k0 + koff);
            acc_i = WMMA_F32(a, bi, acc_i);
            acc_o = WMMA_F32(a, bo, acc_o);
            acc_u = WMMA_F32(a, bu, acc_u);
            // 8 children share B (Wfh tile); unweighted GEMM, weight applied
            // to the accumulator in the epilogue.
#pragma unroll
            for (int k = 0; k < BRANCH; ++k) {
                v2f ah = *(const v2f*)(Hk[k] + k0 + koff);
                acc_fh[k] = WMMA_F32(ah, bf, acc_fh[k]);
            }
        }
    }

    // ---------------- gates + cell update + store -------------------------
    const int j = colbase + l16;               // absolute MEM column
    const float bi_s  = bioux[j]       + biouh[j];
    const float bo_s  = bioux[j + 256] + biouh[j + 256];
    const float bu_s  = bioux[j + 512] + biouh[j + 512];
    const float bfx_s = bfx[j];
    const float bfh_s = bfh[j];
    const size_t outbase = (size_t)node * BM;

#pragma unroll
    for (int v = 0; v < 8; ++v) {
        const int row = rowbase + v + 8 * lhi;   // C/D layout: M = v + 8*(lane/16)
        float iv = sigmoid_f(acc_i[v] + bi_s);
        float ov = sigmoid_f(acc_o[v] + bo_s);
        float uv = tanhf(acc_u[v] + bu_s);
        float c  = iv * uv;
        if (has_children) {
            float fxv = acc_fx[v] + bfx_s;
#pragma unroll
            for (int k = 0; k < BRANCH; ++k) {
                // wgt[k] re-applied here: (w*h)@Wfh^T == w*(h@Wfh^T)
                float f = sigmoid_f(wgt[k] * acc_fh[k][v] + bfh_s + fxv);
                c += f * wgt[k] *
                     c_all[(size_t)(child0 + k) * BM + (size_t)row * MEM + j];
            }
        }
        float h = ov * tanhf(c);
        c_all[outbase + (size_t)row * MEM + j] = c;
        h_all[outbase + (size_t)row * MEM + j] = h;
    }
}

// ---------------------------------------------------------------------------
// Host-side launch: leaves first, then levels 2,1,0 (hsum + node kernel).
// ---------------------------------------------------------------------------
extern "C" void kernel_launch(void* const* d_in, const int* in_sizes, int n_in,
                              void* d_out, int out_size, void* d_ws, size_t ws_size,
                              hipStream_t stream) {
    const float* inputs = (const float*)d_in[0];
    const float* prob   = (const float*)d_in[1];
    const float* Wioux  = (const float*)d_in[2];
    const float* bioux  = (const float*)d_in[3];
    const float* Wiouh  = (const float*)d_in[4];
    const float* biouh  = (const float*)d_in[5];
    const float* Wfx    = (const float*)d_in[6];
    const float* bfx    = (const float*)d_in[7];
    const float* Wfh    = (const float*)d_in[8];
    const float* bfh    = (const float*)d_in[9];

    float* h_all = (float*)d_out;                       // 585 x 256 x 256
    float* c_all = (float*)d_ws;                        // 585 x 256 x 256
    float* hsum  = c_all + (size_t)N_NODES * BM;        // up to 64 x 256 x 256

    const dim3 blk(512);

    // Level 3: leaves, nodes [73, 585), no children.
    tree_lstm_level_kernel<<<dim3(512, 4, 4), blk, 0, stream>>>(
        inputs, prob, Wioux, bioux, Wiouh, biouh, Wfx, bfx, Wfh, bfh,
        hsum, c_all, h_all, 73, 0);

    const int starts[3] = {9, 1, 0};
    const int sizes[3]  = {64, 8, 1};
    for (int l = 0; l < 3; ++l) {
        const int s = starts[l];
        const int L = sizes[l];
        const int total = L * BM;
        hsum_kernel<<<(total + 255) / 256, 256, 0, stream>>>(h_all, prob, hsum, s, L);
        tree_lstm_level_kernel<<<dim3(L, 4, 4), blk, 0, stream>>>(
            inputs, prob, Wioux, bioux, Wiouh, biouh, Wfx, bfx, Wfh, bfh,
            hsum, c_all, h_all, s, 1);
    }
}